// GCN_15092515078148
// MI455X (gfx1250) — compile-verified
//
#include <hip/hip_runtime.h>
#include <hip/hip_bf16.h>

typedef __attribute__((ext_vector_type(2))) float v2f;
typedef __attribute__((ext_vector_type(8))) float v8f;

#define FEAT 128
#define NCLS 7
#define EPSB 1e-5f
#define SCAN_T 1024

// ---------------------------------------------------------------------------
// Zero int scratch (degree counters + scatter cursors).
// ---------------------------------------------------------------------------
__global__ void zero_i32(int* __restrict__ ptr, int n) {
    int i = blockIdx.x * blockDim.x + threadIdx.x;
    if (i < n) ptr[i] = 0;
}

// ---------------------------------------------------------------------------
// deg[row[e]]++  (integer histogram; only 1.6M cheap int atomics)
// ---------------------------------------------------------------------------
__global__ void hist_rows(const int* __restrict__ erow, int* __restrict__ deg, int E) {
    int e = blockIdx.x * blockDim.x + threadIdx.x;
    if (e < E) atomicAdd(&deg[erow[e]], 1);
}

// ---------------------------------------------------------------------------
// Exclusive scan of deg[0..N) -> rowStart[0..N]. Single 1024-thread block:
// each thread serially sums a contiguous chunk, LDS Hillis-Steele scan of the
// 1024 partials, then serial exclusive re-walk.
// ---------------------------------------------------------------------------
__global__ void scan_rows(const int* __restrict__ deg, int* __restrict__ rowStart, int N) {
    __shared__ int part[SCAN_T];
    const int t = threadIdx.x;
    const int chunk = (N + SCAN_T - 1) / SCAN_T;
    const int lo = t * chunk;
    const int hi = (lo + chunk < N) ? lo + chunk : N;
    int s = 0;
    for (int i = lo; i < hi; ++i) s += deg[i];
    part[t] = s;
    __syncthreads();
    for (int off = 1; off < SCAN_T; off <<= 1) {
        int add = (t >= off) ? part[t - off] : 0;
        int v = part[t];
        __syncthreads();
        part[t] = v + add;
        __syncthreads();
    }
    int run = part[t] - s;                 // exclusive prefix of this chunk
    for (int i = lo; i < hi; ++i) { rowStart[i] = run; run += deg[i]; }
    if (t == SCAN_T - 1) rowStart[N] = part[t];
}

// ---------------------------------------------------------------------------
// Counting-sort scatter: edges into row-sorted (scol, sval) arrays.
// ---------------------------------------------------------------------------
__global__ void scatter_edges(const int* __restrict__ erow, const int* __restrict__ ecol,
                              const float* __restrict__ ev, const int* __restrict__ rowStart,
                              int* __restrict__ cursor, int* __restrict__ scol,
                              float* __restrict__ sval, int E) {
    int e = blockIdx.x * blockDim.x + threadIdx.x;
    if (e >= E) return;
    const int r = erow[e];
    const int pos = rowStart[r] + atomicAdd(&cursor[r], 1);
    scol[pos] = ecol[e];
    sval[pos] = ev[e];
}

// ---------------------------------------------------------------------------
// y = x @ W1   (fp32 WMMA 16x16x4, wave32). Block = 128 thr = 4 waves; each
// wave owns a 16-row M tile with the full 128-wide panel (8 C tiles live).
// ---------------------------------------------------------------------------
__global__ void gemm_xW1(const float* __restrict__ X, const float* __restrict__ W,
                         float* __restrict__ Y, int nrows) {
    const int wave = threadIdx.x >> 5;
    const int lane = threadIdx.x & 31;
    const int half = lane >> 4;
    const int r    = lane & 15;
    const int mt   = blockIdx.x * 4 + wave;
    if (mt >= (nrows >> 4)) return;      // wave-uniform exit: EXEC all-1 for WMMA
    const int m0 = mt << 4;

    const v8f vzero = {0.f,0.f,0.f,0.f,0.f,0.f,0.f,0.f};
    v8f acc[8];
#pragma unroll
    for (int nt = 0; nt < 8; ++nt) acc[nt] = vzero;

    const float* arow = X + (size_t)(m0 + r) * FEAT;
    for (int kt = 0; kt < 32; ++kt) {
        const int k0 = kt << 2;
        v2f a;                                       // A: K = 2*half + j, M = r
        a.x = arow[k0 + 2 * half + 0];
        a.y = arow[k0 + 2 * half + 1];
#pragma unroll
        for (int nt = 0; nt < 8; ++nt) {
            const int n0 = nt << 4;
            v2f b;                                   // B: K = 2*half + j, N = r
            b.x = W[(size_t)(k0 + 2 * half + 0) * FEAT + n0 + r];
            b.y = W[(size_t)(k0 + 2 * half + 1) * FEAT + n0 + r];
            acc[nt] = __builtin_amdgcn_wmma_f32_16x16x4_f32(
                false, a, false, b, (short)0, acc[nt], false, false);
        }
    }
#pragma unroll
    for (int nt = 0; nt < 8; ++nt) {                 // C: M = i + 8*half, N = r
        const int n0 = nt << 4;
#pragma unroll
        for (int i = 0; i < 8; ++i)
            Y[(size_t)(m0 + i + 8 * half) * FEAT + n0 + r] = acc[nt][i];
    }
}

// ---------------------------------------------------------------------------
// Fused layer-1 tail: per row r (one wave each):
//   acc = sum_e val * y[col]            (128-wide, float4/lane, CSR, no atomics)
//   h   = relu(bn(acc + b1))            (in registers)
//   p[r,0..6] = h @ W2                  (shfl_xor butterfly reductions)
// z and h are never materialized.
// ---------------------------------------------------------------------------
__global__ void spmm1_fused(const float* __restrict__ y, const int* __restrict__ rowStart,
                            const int* __restrict__ scol, const float* __restrict__ sval,
                            const float* __restrict__ b1, const float* __restrict__ g1,
                            const float* __restrict__ be1, const float* __restrict__ m1,
                            const float* __restrict__ v1, const float* __restrict__ W2,
                            float* __restrict__ p, int N) {
    const int wave = threadIdx.x >> 5;
    const int lane = threadIdx.x & 31;
    const int r = blockIdx.x * 8 + wave;
    if (r >= N) return;
    const int eBeg = rowStart[r];
    const int eEnd = rowStart[r + 1];

    float4 acc = {0.f, 0.f, 0.f, 0.f};
    for (int e = eBeg; e < eEnd; ++e) {
        const int   c = scol[e];
        const float v = sval[e];
        const float4 g = ((const float4*)(y + (size_t)c * FEAT))[lane];
        acc.x = fmaf(v, g.x, acc.x);
        acc.y = fmaf(v, g.y, acc.y);
        acc.z = fmaf(v, g.z, acc.z);
        acc.w = fmaf(v, g.w, acc.w);
    }

    float h[4];
    float af[4] = {acc.x, acc.y, acc.z, acc.w};
#pragma unroll
    for (int j = 0; j < 4; ++j) {
        const int ch = lane * 4 + j;
        const float s = g1[ch] * rsqrtf(v1[ch] + EPSB);
        const float t = (b1[ch] - m1[ch]) * s + be1[ch];
        const float hv = fmaf(af[j], s, t);
        h[j] = hv > 0.0f ? hv : 0.0f;
    }

#pragma unroll
    for (int c = 0; c < NCLS; ++c) {
        float ssum = 0.0f;
#pragma unroll
        for (int j = 0; j < 4; ++j)
            ssum = fmaf(h[j], W2[(size_t)(lane * 4 + j) * NCLS + c], ssum);
#pragma unroll
        for (int off = 16; off >= 1; off >>= 1) ssum += __shfl_xor(ssum, off, 32);
        if (lane == 0) p[(size_t)r * NCLS + c] = ssum;
    }
}

// ---------------------------------------------------------------------------
// Fused layer-2 tail: thread per (row, class):
//   out[r,c] = bn2( sum_e val * p[col, c] + b2 )     (CSR, no atomics)
// ---------------------------------------------------------------------------
__global__ void spmm2_fused(const float* __restrict__ p, const int* __restrict__ rowStart,
                            const int* __restrict__ scol, const float* __restrict__ sval,
                            const float* __restrict__ b2, const float* __restrict__ g2,
                            const float* __restrict__ be2, const float* __restrict__ m2,
                            const float* __restrict__ v2, float* __restrict__ out, int total) {
    int t = blockIdx.x * blockDim.x + threadIdx.x;
    if (t >= total) return;
    const int r = t / NCLS;
    const int c = t - r * NCLS;
    float acc = 0.0f;
    const int eEnd = rowStart[r + 1];
    for (int e = rowStart[r]; e < eEnd; ++e)
        acc = fmaf(sval[e], p[(size_t)scol[e] * NCLS + c], acc);
    const float s = g2[c] * rsqrtf(v2[c] + EPSB);
    out[t] = (acc + b2[c] - m2[c]) * s + be2[c];
}

// ---------------------------------------------------------------------------
extern "C" void kernel_launch(void* const* d_in, const int* in_sizes, int n_in,
                              void* d_out, int out_size, void* d_ws, size_t ws_size,
                              hipStream_t stream) {
    const float* x   = (const float*)d_in[0];
    const float* ev  = (const float*)d_in[1];
    const float* W1  = (const float*)d_in[2];
    const float* b1  = (const float*)d_in[3];
    const float* g1  = (const float*)d_in[4];
    const float* be1 = (const float*)d_in[5];
    const float* m1  = (const float*)d_in[6];
    const float* v1  = (const float*)d_in[7];
    const float* W2  = (const float*)d_in[8];
    const float* b2  = (const float*)d_in[9];
    const float* g2  = (const float*)d_in[10];
    const float* be2 = (const float*)d_in[11];
    const float* m2  = (const float*)d_in[12];
    const float* v2  = (const float*)d_in[13];
    const int*  erow = (const int*)d_in[14];
    const int*  ecol = (const int*)d_in[15];
    float* out = (float*)d_out;

    const int N = in_sizes[0] / FEAT;   // 100000
    const int E = in_sizes[1];          // 1600000

    // Workspace layout:
    //   y [N*128 f32] | p [N*7 f32] | deg [N i32] | cursor [N i32]
    //   | rowStart [N+1 i32] | scol [E i32] | sval [E f32]
    float* y        = (float*)d_ws;
    float* p        = y + (size_t)N * FEAT;
    int*   deg      = (int*)(p + (size_t)N * NCLS);
    int*   cursor   = deg + N;
    int*   rowStart = cursor + N;
    int*   scol     = rowStart + (N + 1);
    float* sval     = (float*)(scol + E);

    // 1) CSR build: zero counters, histogram, scan, scatter (deg+cursor adjacent)
    zero_i32<<<(2 * N + 255) / 256, 256, 0, stream>>>(deg, 2 * N);
    hist_rows<<<(E + 255) / 256, 256, 0, stream>>>(erow, deg, E);
    scan_rows<<<1, SCAN_T, 0, stream>>>(deg, rowStart, N);
    scatter_edges<<<(E + 255) / 256, 256, 0, stream>>>(erow, ecol, ev, rowStart,
                                                       cursor, scol, sval, E);
    // 2) y = x @ W1  (WMMA fp32)
    gemm_xW1<<<((N >> 4) + 3) / 4, 128, 0, stream>>>(x, W1, y, N);
    // 3) p = (relu(bn(SPMM(y) + b1))) @ W2   — fused, no z/h materialization
    spmm1_fused<<<(N + 7) / 8, 256, 0, stream>>>(y, rowStart, scol, sval,
                                                 b1, g1, be1, m1, v1, W2, p, N);
    // 4) out = bn(SPMM(p) + b2)              — fused, writes d_out directly
    spmm2_fused<<<(N * NCLS + 255) / 256, 256, 0, stream>>>(p, rowStart, scol, sval,
                                                            b2, g2, be2, m2, v2, out,
                                                            N * NCLS);
}